// SpatioTemporalGNN_53085795778680
// MI455X (gfx1250) — compile-verified
//
#include <hip/hip_runtime.h>
#include <math.h>

// ---------------------------------------------------------------------------
// SpatioTemporal GNN forward for MI455X (gfx1250, wave32).
// Dense GEMMs use V_WMMA_F32_16X16X4_F32 (full fp32 precision matrix pipe):
//  - templated on K (static unroll, immediate addressing)
//  - 5 M-tiles per wave (80 rows) -> 5 independent WMMA accumulation chains
//  - software-pipelined fragment loads (next k-step prefetched during WMMAs)
// ---------------------------------------------------------------------------

typedef __attribute__((ext_vector_type(2))) float v2f;
typedef __attribute__((ext_vector_type(8))) float v8f;

#define BN_EPS 1e-5f

static constexpr int T_    = 8;
static constexpr int N_    = 50000;     // 625 * 80 -> no M guards needed
static constexpr int E_    = 800000;
static constexpr int INF_  = 32;
static constexpr int MLPF_ = 128;
static constexpr int GCNF_ = 64;
static constexpr int HID2_ = 128;
static constexpr int OUTC_ = 4;

static constexpr int RT_   = 5;         // M-tiles (16 rows each) per wave
static_assert(N_ % (16 * RT_) == 0, "row blocking must divide N");

// ---------------------------------------------------------------------------
// Utility
// ---------------------------------------------------------------------------
__global__ void k_zero(float* __restrict__ p, int n) {
  int i = blockIdx.x * blockDim.x + threadIdx.x;
  if (i < n) p[i] = 0.0f;
}

__global__ void k_fill(float* __restrict__ p, int n, float v) {
  int i = blockIdx.x * blockDim.x + threadIdx.x;
  if (i < n) p[i] = v;
}

// ---------------------------------------------------------------------------
// fp32 WMMA GEMM:  C[M,F] = A[M,K] @ B (+ bias)
//   TRANSB == false : B is row-major [K,F]
//   TRANSB == true  : B is row-major [F,K]  (multiply by B^T, GRU weights)
// One wave computes RT_ stacked 16x16 tiles (80 rows x 16 cols); 4 waves per
// block cover 64 output columns. Fragment layouts per CDNA5 ISA 7.12.2.
// Double-buffered: k-step i+1 fragments load while step i WMMAs execute.
// ---------------------------------------------------------------------------
template <int K, bool TRANSB, bool HAS_BIAS>
__global__ void k_wmma_gemm(const float* __restrict__ A,
                            const float* __restrict__ B,
                            const float* __restrict__ bias,
                            float* __restrict__ C, int F)
{
  const int lane = threadIdx.x & 31;
  const int wave = threadIdx.x >> 5;
  const int half = lane >> 4;       // 0: K pair (0,1) ; 1: K pair (2,3)
  const int l16  = lane & 15;

  const int row0 = blockIdx.x * (16 * RT_);
  const int bcol = (blockIdx.y * 4 + wave) * 16 + l16;

  // Per-lane base pointers (all further addressing is static immediates).
  const float* aBase = A + (size_t)(row0 + l16) * K + half * 2;
  const float* bT    = B + (size_t)bcol * K + half * 2;    // TRANSB path
  const float* bR    = B + (size_t)(half * 2) * F + bcol;  // !TRANSB path

  auto loadB = [&](int k) -> v2f {
    if (TRANSB) {
      return *(const v2f*)(bT + k);                        // contiguous 8B
    } else {
      v2f b;
      b.x = bR[(size_t)(k + 0) * F];
      b.y = bR[(size_t)(k + 1) * F];
      return b;
    }
  };
  auto loadA = [&](int t, int k) -> v2f {
    return *(const v2f*)(aBase + (size_t)t * 16 * K + k);
  };

  v8f acc[RT_];
#pragma unroll
  for (int t = 0; t < RT_; ++t) acc[t] = (v8f){};

  // Prologue: fragments for k = 0
  v2f bC = loadB(0);
  v2f aC[RT_];
#pragma unroll
  for (int t = 0; t < RT_; ++t) aC[t] = loadA(t, 0);

#pragma unroll
  for (int k = 0; k < K; k += 4) {
    v2f bNx = bC;
    v2f aNx[RT_];
    if (k + 4 < K) {                       // statically folded per iteration
      bNx = loadB(k + 4);
#pragma unroll
      for (int t = 0; t < RT_; ++t) aNx[t] = loadA(t, k + 4);
    }
#pragma unroll
    for (int t = 0; t < RT_; ++t) {
      acc[t] = __builtin_amdgcn_wmma_f32_16x16x4_f32(
          /*neg_a=*/false, aC[t], /*neg_b=*/false, bC,
          /*c_mod=*/(short)0, acc[t], /*reuse_a=*/false, /*reuse_b=*/false);
    }
    bC = bNx;
#pragma unroll
    for (int t = 0; t < RT_; ++t)
      if (k + 4 < K) aC[t] = aNx[t];
  }

  const float bi = HAS_BIAS ? bias[bcol] : 0.0f;
  // C/D layout: VGPR i -> M = i (lanes 0-15) / M = 8+i (lanes 16-31), N = l16
#pragma unroll
  for (int t = 0; t < RT_; ++t) {
#pragma unroll
    for (int i = 0; i < 8; ++i) {
      const int r = row0 + t * 16 + half * 8 + i;
      C[(size_t)r * F + bcol] = acc[t][i] + bi;
    }
  }
}

// ---------------------------------------------------------------------------
// BatchNorm (train-mode batch stats over N rows, biased variance)
// ---------------------------------------------------------------------------
__global__ void k_bn_partial(const float* __restrict__ X, int N, int F,
                             float* __restrict__ sums /* [2*F]: sum, sumsq */)
{
  const int f   = threadIdx.x % F;        // F divides 256 (64 or 128)
  const int rpb = blockDim.x / F;
  const int r0  = threadIdx.x / F;
  float s = 0.0f, s2 = 0.0f;
  for (int r = blockIdx.x * rpb + r0; r < N; r += gridDim.x * rpb) {
    const float v = X[(size_t)r * F + f];
    s += v; s2 += v * v;
  }
  atomicAdd(&sums[f], s);
  atomicAdd(&sums[F + f], s2);
}

__global__ void k_bn_finalize(float* __restrict__ sums, int F, float invN)
{
  int f = blockIdx.x * blockDim.x + threadIdx.x;
  if (f < F) {
    const float m = sums[f] * invN;
    const float v = fmaxf(sums[F + f] * invN - m * m, 0.0f);
    sums[f]     = m;
    sums[F + f] = rsqrtf(v + BN_EPS);
  }
}

// y = relu((x - mean) * rstd * g + be);  residual: Y += y else Y = y
__global__ void k_bn_relu_apply(const float* __restrict__ X,
                                const float* __restrict__ sums,
                                const float* __restrict__ g,
                                const float* __restrict__ be,
                                float* __restrict__ Y,
                                int N, int F, int residual)
{
  int i = blockIdx.x * blockDim.x + threadIdx.x;
  if (i >= N * F) return;
  const int f = i % F;
  float a = (X[i] - sums[f]) * sums[F + f] * g[f] + be[f];
  a = fmaxf(a, 0.0f);
  if (residual) Y[i] += a; else Y[i] = a;
}

// ---------------------------------------------------------------------------
// GCN normalization + scatter aggregation
// ---------------------------------------------------------------------------
__global__ void k_deg_edges(const int* __restrict__ dst,
                            const float* __restrict__ ew,
                            float* __restrict__ deg, int E)
{
  int e = blockIdx.x * blockDim.x + threadIdx.x;
  if (e < E) atomicAdd(&deg[dst[e]], ew[e]);
}

__global__ void k_dinv(const float* __restrict__ deg, float* __restrict__ dinv, int n)
{
  int i = blockIdx.x * blockDim.x + threadIdx.x;
  if (i < n) {
    const float d = deg[i];
    dinv[i] = (d > 0.0f) ? rsqrtf(d) : 0.0f;
  }
}

// AGG[n,:] = XW[n,:] * dinv[n]^2   (self-loop term, also zero-initializes AGG)
__global__ void k_agg_self(const float* __restrict__ XW,
                           const float* __restrict__ dinv,
                           float* __restrict__ AGG, int N, int F)
{
  int i = blockIdx.x * blockDim.x + threadIdx.x;
  if (i >= N * F) return;
  const int n = i / F;
  const float dv = dinv[n];
  AGG[i] = XW[i] * (dv * dv);
}

// One wave per edge; each lane moves 2 of the 64 features.
__global__ void k_agg_edges(const int* __restrict__ src,
                            const int* __restrict__ dst,
                            const float* __restrict__ ew,
                            const float* __restrict__ dinv,
                            const float* __restrict__ XW,
                            float* __restrict__ AGG, int E)
{
  const int wpb = blockDim.x >> 5;
  const int e   = blockIdx.x * wpb + (threadIdx.x >> 5);
  if (e >= E) return;
  const int s = src[e];
  const int d = dst[e];
  const float nrm = dinv[s] * ew[e] * dinv[d];
  const int f = (threadIdx.x & 31) * 2;
  v2f v = *(const v2f*)(XW + (size_t)s * GCNF_ + f);
  atomicAdd(&AGG[(size_t)d * GCNF_ + f + 0], v.x * nrm);
  atomicAdd(&AGG[(size_t)d * GCNF_ + f + 1], v.y * nrm);
}

// ---------------------------------------------------------------------------
// GRU gate fusion (gi/gh already contain x@W_ih^T + b_ih and h@W_hh^T + b_hh)
// ---------------------------------------------------------------------------
__global__ void k_gru_gate(const float* __restrict__ gi,
                           const float* __restrict__ gh,
                           float* __restrict__ h, int N, int G)
{
  int i = blockIdx.x * blockDim.x + threadIdx.x;
  if (i >= N * G) return;
  const int n = i / G, f = i % G;
  const float* gin = gi + (size_t)n * 3 * G;
  const float* ghn = gh + (size_t)n * 3 * G;
  const float r  = 1.0f / (1.0f + expf(-(gin[f] + ghn[f])));
  const float z  = 1.0f / (1.0f + expf(-(gin[G + f] + ghn[G + f])));
  const float ng = tanhf(gin[2 * G + f] + r * ghn[2 * G + f]);
  h[i] = (1.0f - z) * ng + z * h[i];
}

// ---------------------------------------------------------------------------
// Final tiny projection: out[N,4] = Z[N,128] @ Wc2[128,4] + bc2
// ---------------------------------------------------------------------------
__global__ void k_final(const float* __restrict__ Z,
                        const float* __restrict__ Wc2,
                        const float* __restrict__ bc2,
                        float* __restrict__ out, int N)
{
  int i = blockIdx.x * blockDim.x + threadIdx.x;
  if (i >= N * OUTC_) return;
  const int n = i / OUTC_, c = i % OUTC_;
  float s = bc2[c];
  const float* z = Z + (size_t)n * HID2_;
#pragma unroll 8
  for (int k = 0; k < HID2_; ++k) s += z[k] * Wc2[k * OUTC_ + c];
  out[i] = s;
}

// ---------------------------------------------------------------------------
// Host-side GEMM dispatch (static K)
// ---------------------------------------------------------------------------
template <int K>
static void gemm_launch(const float* A, const float* B, const float* bias,
                        float* C, int F, int transB, hipStream_t stream)
{
  dim3 grid(N_ / (16 * RT_), F / 64);
  dim3 blk(128);
  if (transB) {
    if (bias) k_wmma_gemm<K, true,  true ><<<grid, blk, 0, stream>>>(A, B, bias, C, F);
    else      k_wmma_gemm<K, true,  false><<<grid, blk, 0, stream>>>(A, B, nullptr, C, F);
  } else {
    if (bias) k_wmma_gemm<K, false, true ><<<grid, blk, 0, stream>>>(A, B, bias, C, F);
    else      k_wmma_gemm<K, false, false><<<grid, blk, 0, stream>>>(A, B, nullptr, C, F);
  }
}

// ---------------------------------------------------------------------------
// Orchestration
// ---------------------------------------------------------------------------
extern "C" void kernel_launch(void* const* d_in, const int* in_sizes, int n_in,
                              void* d_out, int out_size, void* d_ws, size_t ws_size,
                              hipStream_t stream)
{
  (void)in_sizes; (void)n_in; (void)out_size; (void)ws_size;

  const float* xs   = (const float*)d_in[0];
  const int*   esrc = (const int*)  d_in[1];
  const int*   edst = (const int*)  d_in[2];
  const float* ewt  = (const float*)d_in[3];
  const float* W1   = (const float*)d_in[4];
  // b1 (d_in[5]) cancels under BN (mean subtraction), as do b2/gb/bp/bc1
  const float* g1   = (const float*)d_in[6];
  const float* be1  = (const float*)d_in[7];
  const float* W2   = (const float*)d_in[8];
  const float* g2   = (const float*)d_in[10];
  const float* be2  = (const float*)d_in[11];
  const float* gW   = (const float*)d_in[12];
  const float* gg   = (const float*)d_in[14];
  const float* gbe  = (const float*)d_in[15];
  const float* Wp   = (const float*)d_in[16];
  const float* gp   = (const float*)d_in[18];
  const float* bep  = (const float*)d_in[19];
  const float* W_ih = (const float*)d_in[20];
  const float* W_hh = (const float*)d_in[21];
  const float* b_ih = (const float*)d_in[22];
  const float* b_hh = (const float*)d_in[23];
  const float* Wc1  = (const float*)d_in[24];
  const float* gc   = (const float*)d_in[26];
  const float* bec  = (const float*)d_in[27];
  const float* Wc2  = (const float*)d_in[28];
  const float* bc2  = (const float*)d_in[29];

  // Workspace layout (fp32), ~167 MB total -> mostly resident in 192 MB L2
  float* ws = (float*)d_ws;
  float* X1   = ws; ws += (size_t)N_ * MLPF_;      // [N,128] MLP1 / classifier scratch
  float* H    = ws; ws += (size_t)N_ * GCNF_;      // [N,64] running node state
  float* XW   = ws; ws += (size_t)N_ * GCNF_;      // [N,64] GEMM temp
  float* AGG  = ws; ws += (size_t)N_ * GCNF_;      // [N,64] aggregation temp
  float* H2   = ws; ws += (size_t)N_ * GCNF_;      // [N,64] encoder output per t
  float* HS   = ws; ws += (size_t)N_ * GCNF_;      // [N,64] GRU hidden state
  float* GI   = ws; ws += (size_t)N_ * 3 * GCNF_;  // [N,192]
  float* GH   = ws; ws += (size_t)N_ * 3 * GCNF_;  // [N,192]
  float* DEG  = ws; ws += N_;
  float* DINV = ws; ws += N_;
  float* SUMS = ws; ws += 512;                     // BN mean/rstd scratch

  auto bn = [&](const float* X, const float* g, const float* be,
                float* Y, int F, int residual) {
    k_zero<<<1, 256, 0, stream>>>(SUMS, 2 * F);
    k_bn_partial<<<1024, 256, 0, stream>>>(X, N_, F, SUMS);
    k_bn_finalize<<<1, 256, 0, stream>>>(SUMS, F, 1.0f / (float)N_);
    const int tot = N_ * F;
    k_bn_relu_apply<<<(tot + 255) / 256, 256, 0, stream>>>(X, SUMS, g, be, Y,
                                                           N_, F, residual);
  };

  // GRU h0 = 0
  k_zero<<<((N_ * GCNF_) + 255) / 256, 256, 0, stream>>>(HS, N_ * GCNF_);

  for (int t = 0; t < T_; ++t) {
    const float* x_t   = xs   + (size_t)t * N_ * INF_;
    const int*   src_t = esrc + (size_t)t * E_;
    const int*   dst_t = edst + (size_t)t * E_;
    const float* ew_t  = ewt  + (size_t)t * E_;

    // pre-MLP
    gemm_launch<INF_>(x_t, W1, nullptr, X1, MLPF_, 0, stream);
    bn(X1, g1, be1, X1, MLPF_, 0);
    gemm_launch<MLPF_>(X1, W2, nullptr, XW, GCNF_, 0, stream);
    bn(XW, g2, be2, H, GCNF_, 0);

    // degree / normalization (shared by all 3 GCN layers at this t)
    k_fill<<<(N_ + 255) / 256, 256, 0, stream>>>(DEG, N_, 1.0f);  // self-loop w=1
    k_deg_edges<<<(E_ + 255) / 256, 256, 0, stream>>>(dst_t, ew_t, DEG, E_);
    k_dinv<<<(N_ + 255) / 256, 256, 0, stream>>>(DEG, DINV, N_);

    // 3 GCN blocks with residual
    for (int l = 0; l < 3; ++l) {
      gemm_launch<GCNF_>(H, gW + (size_t)l * GCNF_ * GCNF_, nullptr, XW,
                         GCNF_, 0, stream);
      k_agg_self<<<((N_ * GCNF_) + 255) / 256, 256, 0, stream>>>(XW, DINV, AGG,
                                                                 N_, GCNF_);
      k_agg_edges<<<(E_ + 7) / 8, 256, 0, stream>>>(src_t, dst_t, ew_t, DINV,
                                                    XW, AGG, E_);
      bn(AGG, gg + l * GCNF_, gbe + l * GCNF_, H, GCNF_, /*residual=*/1);
    }

    // post-MLP
    gemm_launch<GCNF_>(H, Wp, nullptr, XW, GCNF_, 0, stream);
    bn(XW, gp, bep, H2, GCNF_, 0);

    // GRU step (interleaved with encoder: scan consumes H sequentially)
    gemm_launch<GCNF_>(H2, W_ih, b_ih, GI, 3 * GCNF_, 1, stream);
    gemm_launch<GCNF_>(HS, W_hh, b_hh, GH, 3 * GCNF_, 1, stream);
    k_gru_gate<<<((N_ * GCNF_) + 255) / 256, 256, 0, stream>>>(GI, GH, HS,
                                                               N_, GCNF_);
  }

  // classifier
  gemm_launch<GCNF_>(HS, Wc1, nullptr, X1, HID2_, 0, stream);
  bn(X1, gc, bec, X1, HID2_, 0);
  k_final<<<((N_ * OUTC_) + 255) / 256, 256, 0, stream>>>(X1, Wc2, bc2,
                                                          (float*)d_out, N_);
}